// MSE_OHEM_Loss_3169685865353
// MI455X (gfx1250) — compile-verified
//
#include <hip/hip_runtime.h>
#include <hip/hip_bf16.h>
#include <stdint.h>

// ---------------------------------------------------------------------------
// MSE-OHEM loss for MI455X (gfx1250).
// Memory-bound op: 151 MB inputs, ~4 streaming passes (working set fits in
// 192MB L2 after pass 1). Exact top-k via 3-level radix-select on float bit
// patterns (integer-atomic histograms only -> deterministic). Async
// global->LDS (ASYNCcnt path, gfx1250) feeds the two heavy compute passes.
// ---------------------------------------------------------------------------

#define BATCH 64
#define CHW   (2 * 384 * 384)        // 294912 elements per sample
#define TPB   256                    // 8 wave32 per block
#define TILE  4096                   // floats per block tile
#define NBLK  (CHW / TILE)           // 72 blocks per sample (exact)

typedef int v4i_t __attribute__((ext_vector_type(4)));
typedef __attribute__((address_space(1))) v4i_t gv4i_t;  // global
typedef __attribute__((address_space(3))) v4i_t lv4i_t;  // LDS

struct Ws {
  // integer-atomic accumulated regions (must be zeroed every launch):
  unsigned hist1[BATCH][1024];   // float bits [30:21]  (bit31==0, loss>=0)
  unsigned hist2[BATCH][2048];   // float bits [20:10]
  unsigned hist3[BATCH][1024];   // float bits [9:0]
  unsigned numPos[BATCH];
  // fully-overwritten state:
  unsigned kArr[BATCH];
  unsigned b1[BATCH], b2[BATCH];
  unsigned cnt1[BATCH], cnt2[BATCH];
  unsigned countGt[BATCH];
  unsigned thrBits[BATCH];
  float blkTot[BATCH][NBLK];
  float blkPos[BATCH][NBLK];
  float blkTopk[BATCH][NBLK];
  float totalSum[BATCH], posSum[BATCH];
};
#define ZERO_WORDS (BATCH * 1024 + BATCH * 2048 + BATCH * 1024 + BATCH)

// --------------------------- async tile helpers ----------------------------
// Device pass: unconditional use of the gfx1250 async-copy builtin (verified
// present in round 1 via its parameter-type diagnostic). If this compiles,
// global_load_async_to_lds_b128 is guaranteed to be in the object.

__device__ __forceinline__ void tile_copy_async(const float* __restrict__ g,
                                                float* l, int tid) {
#if defined(__HIP_DEVICE_COMPILE__)
  gv4i_t* gp = (gv4i_t*)g;   // addrspacecast + drops const
  lv4i_t* lp = (lv4i_t*)l;
#pragma unroll
  for (int q = 0; q < TILE / (TPB * 4); ++q) {
    int f = q * TPB + tid;   // one b128 (4 floats) per lane per step
    __builtin_amdgcn_global_load_async_to_lds_b128(gp + f, lp + f, 0, 0);
  }
#else
  // host-pass parse only
#pragma unroll
  for (int q = 0; q < TILE / (TPB * 4); ++q) {
    int f = q * TPB + tid;
    reinterpret_cast<float4*>(l)[f] = reinterpret_cast<const float4*>(g)[f];
  }
#endif
}

__device__ __forceinline__ void tiles_ready() {
#if defined(__HIP_DEVICE_COMPILE__)
#if __has_builtin(__builtin_amdgcn_s_wait_asynccnt)
  __builtin_amdgcn_s_wait_asynccnt(0);
#else
  asm volatile("s_wait_asynccnt 0" ::: "memory");
#endif
#endif
  __syncthreads();
}

// ------------------------------- kernels -----------------------------------

__global__ void ohem_zero(unsigned* p, int n) {
  int i = blockIdx.x * blockDim.x + threadIdx.x;
  if (i < n) p[i] = 0u;
}

// Pass 1: loss stats (total sum, pos sum, num_pos) + level-1 histogram.
__global__ void __launch_bounds__(TPB) ohem_pass1(const float* __restrict__ o,
                                                  const float* __restrict__ t,
                                                  Ws* ws) {
  __shared__ float shO[TILE];
  __shared__ float shT[TILE];
  __shared__ unsigned shHist[1024];
  __shared__ float red[TPB];
  __shared__ unsigned shNp;
  const int tid = threadIdx.x;
  const int s = blockIdx.y;
  const int bx = blockIdx.x;
  const size_t base = (size_t)s * CHW + (size_t)bx * TILE;

  for (int i = tid; i < 1024; i += TPB) shHist[i] = 0u;
  if (tid == 0) shNp = 0u;
  tile_copy_async(o + base, shO, tid);
  tile_copy_async(t + base, shT, tid);
  tiles_ready();

  float tot = 0.f, ps = 0.f;
  unsigned np = 0u;
#pragma unroll
  for (int j = 0; j < TILE / TPB; ++j) {
    int e = j * TPB + tid;
    float tv = shT[e];
    float d = shO[e] - tv;
    float loss = d * d;
    tot += loss;
    if (tv > 0.f) {
      ps += loss;
      ++np;
    } else {
      atomicAdd(&shHist[__float_as_uint(loss) >> 21], 1u);
    }
  }
  // deterministic tree reductions (fixed pairing order)
  red[tid] = tot;
  __syncthreads();
  for (int off = TPB / 2; off > 0; off >>= 1) {
    if (tid < off) red[tid] += red[tid + off];
    __syncthreads();
  }
  if (tid == 0) ws->blkTot[s][bx] = red[0];
  __syncthreads();
  red[tid] = ps;
  __syncthreads();
  for (int off = TPB / 2; off > 0; off >>= 1) {
    if (tid < off) red[tid] += red[tid + off];
    __syncthreads();
  }
  if (tid == 0) ws->blkPos[s][bx] = red[0];
  atomicAdd(&shNp, np);
  __syncthreads();
  if (tid == 0) atomicAdd(&ws->numPos[s], shNp);
  for (int i = tid; i < 1024; i += TPB) {
    unsigned c = shHist[i];
    if (c) atomicAdd(&ws->hist1[s][i], c);
  }
}

// Select 1: serial (deterministic) per-sample reduce + coarse bin selection.
__global__ void ohem_select1(Ws* ws) {
  int s = threadIdx.x;
  if (s >= BATCH) return;
  float tot = 0.f, ps = 0.f;
  for (int i = 0; i < NBLK; ++i) {
    tot += ws->blkTot[s][i];
    ps += ws->blkPos[s][i];
  }
  ws->totalSum[s] = tot;
  ws->posSum[s] = ps;
  unsigned np = ws->numPos[s];
  unsigned neg = (unsigned)CHW - np;
  unsigned k = (3u * np < neg) ? 3u * np : neg;
  ws->kArr[s] = k;
  if (k >= 10u) {
    unsigned c = 0u;
    int b = 1023;
    for (; b > 0; --b) {
      unsigned h = ws->hist1[s][b];
      if (c + h >= k) break;
      c += h;
    }
    ws->b1[s] = (unsigned)b;
    ws->cnt1[s] = c;  // count strictly above bin b
  }
}

// Pass 2: level-2 histogram (elements in the selected coarse bin only).
__global__ void __launch_bounds__(TPB) ohem_pass2(const float* __restrict__ o,
                                                  const float* __restrict__ t,
                                                  Ws* ws) {
  const int s = blockIdx.y;
  const unsigned k = ws->kArr[s];
  if (k < 10u) return;  // uniform per-block exit, barriers safe
  const unsigned b1v = ws->b1[s];
  __shared__ unsigned shHist[2048];
  const int tid = threadIdx.x;
  for (int i = tid; i < 2048; i += TPB) shHist[i] = 0u;
  __syncthreads();
  const size_t base = (size_t)s * CHW + (size_t)blockIdx.x * TILE;
  const float4* o4 = reinterpret_cast<const float4*>(o + base);
  const float4* t4 = reinterpret_cast<const float4*>(t + base);
#pragma unroll
  for (int q = 0; q < TILE / (TPB * 4); ++q) {
    float4 ov = o4[q * TPB + tid];
    float4 tv = t4[q * TPB + tid];
    float ox[4] = {ov.x, ov.y, ov.z, ov.w};
    float tx[4] = {tv.x, tv.y, tv.z, tv.w};
#pragma unroll
    for (int c = 0; c < 4; ++c) {
      if (!(tx[c] > 0.f)) {
        float d = ox[c] - tx[c];
        unsigned bits = __float_as_uint(d * d);
        if ((bits >> 21) == b1v) atomicAdd(&shHist[(bits >> 10) & 2047u], 1u);
      }
    }
  }
  __syncthreads();
  for (int i = tid; i < 2048; i += TPB) {
    unsigned c = shHist[i];
    if (c) atomicAdd(&ws->hist2[s][i], c);
  }
}

__global__ void ohem_select2(Ws* ws) {
  int s = threadIdx.x;
  if (s >= BATCH) return;
  unsigned k = ws->kArr[s];
  if (k < 10u) return;
  unsigned kk = k - ws->cnt1[s];
  unsigned c = 0u;
  int b = 2047;
  for (; b > 0; --b) {
    unsigned h = ws->hist2[s][b];
    if (c + h >= kk) break;
    c += h;
  }
  ws->b2[s] = (unsigned)b;
  ws->cnt2[s] = ws->cnt1[s] + c;
}

// Pass 3: level-3 histogram (exact low bits).
__global__ void __launch_bounds__(TPB) ohem_pass3(const float* __restrict__ o,
                                                  const float* __restrict__ t,
                                                  Ws* ws) {
  const int s = blockIdx.y;
  const unsigned k = ws->kArr[s];
  if (k < 10u) return;
  const unsigned b1v = ws->b1[s];
  const unsigned b2v = ws->b2[s];
  __shared__ unsigned shHist[1024];
  const int tid = threadIdx.x;
  for (int i = tid; i < 1024; i += TPB) shHist[i] = 0u;
  __syncthreads();
  const size_t base = (size_t)s * CHW + (size_t)blockIdx.x * TILE;
  const float4* o4 = reinterpret_cast<const float4*>(o + base);
  const float4* t4 = reinterpret_cast<const float4*>(t + base);
#pragma unroll
  for (int q = 0; q < TILE / (TPB * 4); ++q) {
    float4 ov = o4[q * TPB + tid];
    float4 tv = t4[q * TPB + tid];
    float ox[4] = {ov.x, ov.y, ov.z, ov.w};
    float tx[4] = {tv.x, tv.y, tv.z, tv.w};
#pragma unroll
    for (int c = 0; c < 4; ++c) {
      if (!(tx[c] > 0.f)) {
        float d = ox[c] - tx[c];
        unsigned bits = __float_as_uint(d * d);
        if ((bits >> 21) == b1v && ((bits >> 10) & 2047u) == b2v)
          atomicAdd(&shHist[bits & 1023u], 1u);
      }
    }
  }
  __syncthreads();
  for (int i = tid; i < 1024; i += TPB) {
    unsigned c = shHist[i];
    if (c) atomicAdd(&ws->hist3[s][i], c);
  }
}

__global__ void ohem_select3(Ws* ws) {
  int s = threadIdx.x;
  if (s >= BATCH) return;
  unsigned k = ws->kArr[s];
  if (k < 10u) return;
  unsigned kk = k - ws->cnt2[s];
  unsigned c = 0u;
  int b = 1023;
  for (; b > 0; --b) {
    unsigned h = ws->hist3[s][b];
    if (c + h >= kk) break;
    c += h;
  }
  ws->countGt[s] = ws->cnt2[s] + c;  // count strictly greater than threshold
  ws->thrBits[s] = (ws->b1[s] << 21) | (ws->b2[s] << 10) | (unsigned)b;
}

// Pass 4: sum of negative losses strictly above threshold (per-block partials).
__global__ void __launch_bounds__(TPB) ohem_pass4(const float* __restrict__ o,
                                                  const float* __restrict__ t,
                                                  Ws* ws) {
  const int s = blockIdx.y;
  const unsigned k = ws->kArr[s];
  if (k < 10u) return;
  const unsigned thr = ws->thrBits[s];
  __shared__ float shO[TILE];
  __shared__ float shT[TILE];
  __shared__ float red[TPB];
  const int tid = threadIdx.x;
  const int bx = blockIdx.x;
  const size_t base = (size_t)s * CHW + (size_t)bx * TILE;
  tile_copy_async(o + base, shO, tid);
  tile_copy_async(t + base, shT, tid);
  tiles_ready();
  float acc = 0.f;
#pragma unroll
  for (int j = 0; j < TILE / TPB; ++j) {
    int e = j * TPB + tid;
    float tv = shT[e];
    if (!(tv > 0.f)) {
      float d = shO[e] - tv;
      float loss = d * d;
      if (__float_as_uint(loss) > thr) acc += loss;
    }
  }
  red[tid] = acc;
  __syncthreads();
  for (int off = TPB / 2; off > 0; off >>= 1) {
    if (tid < off) red[tid] += red[tid + off];
    __syncthreads();
  }
  if (tid == 0) ws->blkTopk[s][bx] = red[0];
}

// Finalize: per-sample combine + batch mean (all fixed-order float math).
__global__ void ohem_finalize(Ws* ws, float* out) {
  __shared__ float red[BATCH];
  int s = threadIdx.x;
  float per = 0.f;
  if (s < BATCH) {
    unsigned k = ws->kArr[s];
    if (k < 10u) {
      per = ws->totalSum[s] / (float)CHW;
    } else {
      float sumGt = 0.f;
      for (int i = 0; i < NBLK; ++i) sumGt += ws->blkTopk[s][i];
      unsigned rem = k - ws->countGt[s];  // ties at threshold value
      float v = __uint_as_float(ws->thrBits[s]);
      float topk = sumGt + (float)rem * v;
      unsigned np = ws->numPos[s];
      if (np == 0u) np = 1u;
      per = ws->posSum[s] / (float)np + topk / (float)k;
    }
  }
  red[s] = per;
  __syncthreads();
  for (int off = BATCH / 2; off > 0; off >>= 1) {
    if (s < off) red[s] += red[s + off];
    __syncthreads();
  }
  if (s == 0) out[0] = red[0] / (float)BATCH;
}

// ------------------------------- launcher ----------------------------------

extern "C" void kernel_launch(void* const* d_in, const int* in_sizes, int n_in,
                              void* d_out, int out_size, void* d_ws, size_t ws_size,
                              hipStream_t stream) {
  (void)in_sizes; (void)n_in; (void)out_size; (void)ws_size;
  const float* o = (const float*)d_in[0];
  const float* t = (const float*)d_in[1];
  float* out = (float*)d_out;
  Ws* ws = (Ws*)d_ws;

  ohem_zero<<<(ZERO_WORDS + TPB - 1) / TPB, TPB, 0, stream>>>((unsigned*)d_ws, ZERO_WORDS);
  dim3 grid(NBLK, BATCH);
  ohem_pass1<<<grid, TPB, 0, stream>>>(o, t, ws);
  ohem_select1<<<1, BATCH, 0, stream>>>(ws);
  ohem_pass2<<<grid, TPB, 0, stream>>>(o, t, ws);
  ohem_select2<<<1, BATCH, 0, stream>>>(ws);
  ohem_pass3<<<grid, TPB, 0, stream>>>(o, t, ws);
  ohem_select3<<<1, BATCH, 0, stream>>>(ws);
  ohem_pass4<<<grid, TPB, 0, stream>>>(o, t, ws);
  ohem_finalize<<<1, BATCH, 0, stream>>>(ws, out);
}